// StockLSTM_27195732918374
// MI455X (gfx1250) — compile-verified
//
#include <hip/hip_runtime.h>
#include <hip/hip_bf16.h>

// ---------------------------------------------------------------------------
// Fused LSTM (BATCH=8192, SEQ=512, IN=5, HID=64) for gfx1250 (MI455X).
//
// Roofline: recurrent GEMM = 137 GFLOP fp32, sequential over 512 steps.
// HBM traffic if fused = 84 MB (inputs) + 66 KB (weights) + 32 KB (out)
// => ~4 us at 23.3 TB/s: compute-bound. We therefore fuse the entire scan
// into one kernel: h/c live in VGPRs, weights live in LDS, and the per-step
// GEMM runs on V_WMMA_F32_16X16X4_F32 (full fp32 precision for the
// numerically-sensitive recurrence).
//
// Trick: gates = [h | x | 1 | pad] (K=72) x [W_hh | W_ih | b_ih+b_hh | 0]^T,
// so bias + input projection are folded into the same WMMA chain and the
// accumulator starts at zero each step.
// ---------------------------------------------------------------------------

typedef __attribute__((ext_vector_type(2))) float v2f;
typedef __attribute__((ext_vector_type(8))) float v8f;

#define BATCH   8192
#define SEQLEN  512
#define NIN     5
#define HID     64
#define NGATE   256        // 4*HID
#define KTOT    72         // 64 (h) + 5 (x) + 1 (bias one) + 2 pad
#define NCHUNK  18         // KTOT / 4
#define WSTR    76         // LDS row stride for packed W  (76 % 64 == 12 -> conflict-free b64)
#define HSTR    68         // LDS row stride for h bounce  (68 % 64 == 4  -> conflict-free)
#define WAVES   4
#define TPB     (WAVES * 32)
#define ROWS_PER_WAVE  16
#define ROWS_PER_BLOCK (WAVES * ROWS_PER_WAVE)
#define LDS_W_FLOATS   (NGATE * WSTR)
#define LDS_H_FLOATS   (16 * HSTR)
#define LDS_BYTES      ((LDS_W_FLOATS + WAVES * LDS_H_FLOATS) * 4)

__device__ __forceinline__ float fsig(float x) {
    return 1.0f / (1.0f + __expf(-x));          // v_exp_f32 path
}
__device__ __forceinline__ float ftanh(float x) {
    float e = __expf(-2.0f * x);
    return (1.0f - e) / (1.0f + e);
}

__global__ __launch_bounds__(TPB) void
StockLSTM_fused_wmma(const float* __restrict__ inputs,   // [B, S, IN]
                     const float* __restrict__ W_ih,     // [4H, IN]
                     const float* __restrict__ W_hh,     // [4H, H]
                     const float* __restrict__ b_ih,     // [4H]
                     const float* __restrict__ b_hh,     // [4H]
                     const float* __restrict__ fc_w,     // [1, H]
                     const float* __restrict__ fc_b,     // [1]
                     float* __restrict__ out)            // [B]
{
    extern __shared__ float lds[];
    float* Wlds = lds;                                   // [NGATE][WSTR]
    const int tid  = threadIdx.x;
    const int wave = tid >> 5;
    const int lane = tid & 31;
    float* hbuf = lds + LDS_W_FLOATS + wave * LDS_H_FLOATS;   // per-wave [16][HSTR]

    // ---- Stage packed weight matrix into LDS:  W[g][k] over K=72 ----
    for (int idx = tid; idx < LDS_W_FLOATS; idx += TPB) {
        int g = idx / WSTR;
        int k = idx - g * WSTR;
        float v = 0.0f;
        if (k < HID)                v = W_hh[g * HID + k];
        else if (k < HID + NIN)     v = W_ih[g * NIN + (k - HID)];
        else if (k == HID + NIN)    v = b_ih[g] + b_hh[g];
        /* k in [70,76): zero (pad; k>=72 never read by fragments) */
        Wlds[idx] = v;
    }
    __syncthreads();

    const int rowBase = blockIdx.x * ROWS_PER_BLOCK + wave * ROWS_PER_WAVE;
    const int m  = lane & 15;        // batch row within the 16-row tile (A-frag M)
    const int dl = lane >> 4;        // lane half: selects K-pair within a chunk
    const float* xrow = inputs + (size_t)(rowBase + m) * SEQLEN * NIN;

    // A fragments: chunks 0..15 = h (starts at 0), chunks 16..17 = [x | 1 | pad]
    v2f afrag[NCHUNK];
#pragma unroll
    for (int c = 0; c < 16; ++c) afrag[c] = (v2f){0.0f, 0.0f};

    float cstate[4][8];
#pragma unroll
    for (int j = 0; j < 4; ++j)
#pragma unroll
        for (int r = 0; r < 8; ++r) cstate[j][r] = 0.0f;

    for (int s = 0; s < SEQLEN; ++s) {
        const float* x = xrow + (size_t)s * NIN;
        __builtin_prefetch(x + 16 * NIN, 0, 0);          // global_prefetch_b8 ahead

        // chunk 16: K=64..67  ->  lanes 0-15: x0,x1 ; lanes 16-31: x2,x3
        // chunk 17: K=68..71  ->  lanes 0-15: x4,1.0; lanes 16-31: 0,0
        afrag[16] = (v2f){x[2 * dl], x[2 * dl + 1]};
        afrag[17] = dl ? (v2f){0.0f, 0.0f} : (v2f){x[4], 1.0f};

        // Four hid-column groups; tiles {j, j+4, j+8, j+12} = i,f,g,o for the
        // same (m, hid) lane/VGPR positions -> elementwise cell update is local.
#pragma unroll
        for (int j = 0; j < 4; ++j) {
            v8f ai = {}, af = {}, ag = {}, ao = {};
            const int gi = (16 * j +   0 + m) * WSTR;
            const int gf = (16 * j +  64 + m) * WSTR;
            const int gg = (16 * j + 128 + m) * WSTR;
            const int go = (16 * j + 192 + m) * WSTR;
#pragma unroll
            for (int c = 0; c < NCHUNK; ++c) {
                const int k = 4 * c + 2 * dl;            // even -> 8B aligned
                v2f b0 = *(const v2f*)&Wlds[gi + k];
                v2f b1 = *(const v2f*)&Wlds[gf + k];
                v2f b2 = *(const v2f*)&Wlds[gg + k];
                v2f b3 = *(const v2f*)&Wlds[go + k];
                v2f a  = afrag[c];
                ai = __builtin_amdgcn_wmma_f32_16x16x4_f32(false, a, false, b0,
                                                           (short)0, ai, false, false);
                af = __builtin_amdgcn_wmma_f32_16x16x4_f32(false, a, false, b1,
                                                           (short)0, af, false, false);
                ag = __builtin_amdgcn_wmma_f32_16x16x4_f32(false, a, false, b2,
                                                           (short)0, ag, false, false);
                ao = __builtin_amdgcn_wmma_f32_16x16x4_f32(false, a, false, b3,
                                                           (short)0, ao, false, false);
            }
            // LSTM cell update (PyTorch gate order i,f,g,o)
#pragma unroll
            for (int r = 0; r < 8; ++r) {
                float ig = fsig(ai[r]);
                float fg = fsig(af[r]);
                float gc = ftanh(ag[r]);
                float og = fsig(ao[r]);
                float cc = fg * cstate[j][r] + ig * gc;
                cstate[j][r] = cc;
                float hh = og * ftanh(cc);
                // D layout: (M = r + 8*dl, N = 16*j + m) -> write h to bounce buf
                hbuf[(r + 8 * dl) * HSTR + 16 * j + m] = hh;
            }
        }
        // D-layout -> A-fragment relayout via per-wave LDS bounce
        // (same-wave DS ops are in-order; no barrier needed)
#pragma unroll
        for (int c = 0; c < 16; ++c)
            afrag[c] = *(const v2f*)&hbuf[m * HSTR + 4 * c + 2 * dl];
    }

    // ---- Final projection: pred = h . fc_w + fc_b, leaky_relu(0.01) ----
    float partial = 0.0f;
#pragma unroll
    for (int k = 0; k < 32; ++k) {
        int hid = 32 * dl + k;
        partial += hbuf[m * HSTR + hid] * fc_w[hid];
    }
    partial += __shfl_xor(partial, 16, 32);
    if (dl == 0) {
        float p = partial + fc_b[0];
        out[rowBase + m] = p > 0.0f ? p : 0.01f * p;
    }
}

extern "C" void kernel_launch(void* const* d_in, const int* in_sizes, int n_in,
                              void* d_out, int out_size, void* d_ws, size_t ws_size,
                              hipStream_t stream) {
    (void)in_sizes; (void)n_in; (void)d_ws; (void)ws_size; (void)out_size;
    const float* inputs = (const float*)d_in[0];
    const float* W_ih   = (const float*)d_in[1];
    const float* W_hh   = (const float*)d_in[2];
    const float* b_ih   = (const float*)d_in[3];
    const float* b_hh   = (const float*)d_in[4];
    const float* fc_w   = (const float*)d_in[5];
    const float* fc_b   = (const float*)d_in[6];
    float* out = (float*)d_out;

    // Allow >64KB dynamic LDS (CDNA5 WGP supports up to 320KB per workgroup).
    (void)hipFuncSetAttribute((const void*)StockLSTM_fused_wmma,
                              hipFuncAttributeMaxDynamicSharedMemorySize,
                              LDS_BYTES);

    dim3 grid(BATCH / ROWS_PER_BLOCK);   // 128 blocks x 4 waves x 16 rows = 8192
    dim3 block(TPB);
    StockLSTM_fused_wmma<<<grid, block, LDS_BYTES, stream>>>(
        inputs, W_ih, W_hh, b_ih, b_hh, fc_w, fc_b, out);
}